// MPNN_44349832298684
// MI455X (gfx1250) — compile-verified
//
#include <hip/hip_runtime.h>
#include <cstddef>

// -----------------------------------------------------------------------------
// 2-layer GCN for MI455X (gfx1250, wave32, WMMA).
//
// Key reduction: reference gcn_conv scatters h[src]*w to SRC buckets, so it is
// exactly h[i] * c[i] with c[i] = rsqrt(deg[i])*T[i] + 1/deg[i],
// T[i] = sum over edges (src=i, src!=dst) of rsqrt(deg[dst]).
//
// Pipeline: init -> deg atomics -> T atomics -> c -> GEMM1(x@W1^T)*c+b1 with
// fused batchnorm-stat accumulation -> finalize mean/rstd -> GEMM2 with fused
// normalize+relu, *c + b2.
//
// GEMMs use V_WMMA_F32_16X16X4_F32 (fp32 precision ~ reference), tiles staged
// in dynamic LDS, 132-float row stride => conflict-free 64-bit fragment reads.
// -----------------------------------------------------------------------------

typedef __attribute__((ext_vector_type(2))) float v2f;
typedef __attribute__((ext_vector_type(8))) float v8f;

#define LDS_STRIDE 132
#define FEAT 128
#define OUTF 64

// ---------------------------------------------------------------- init
__global__ void gcn_init_kernel(float* __restrict__ deg, float* __restrict__ Tac,
                                float* __restrict__ gsum, float* __restrict__ gssq,
                                int Nn) {
  int i = blockIdx.x * blockDim.x + threadIdx.x;
  if (i < Nn) { deg[i] = 1.0f; Tac[i] = 0.0f; }
  if (i < FEAT) { gsum[i] = 0.0f; gssq[i] = 0.0f; }
}

// ---------------------------------------------------------------- degree
__global__ void gcn_deg_kernel(const int* __restrict__ src, const int* __restrict__ dst,
                               float* __restrict__ deg, int Ecnt) {
  int e = blockIdx.x * blockDim.x + threadIdx.x;
  if (e < Ecnt) {
    int s = src[e], d = dst[e];
    if (s != d) atomicAdd(&deg[d], 1.0f);
  }
}

// ---------------------------------------------------------------- T sums
__global__ void gcn_tsum_kernel(const int* __restrict__ src, const int* __restrict__ dst,
                                const float* __restrict__ deg, float* __restrict__ Tac,
                                int Ecnt) {
  int e = blockIdx.x * blockDim.x + threadIdx.x;
  if (e < Ecnt) {
    int s = src[e], d = dst[e];
    if (s != d) atomicAdd(&Tac[s], rsqrtf(deg[d]));
  }
}

// ---------------------------------------------------------------- c scale
__global__ void gcn_cscale_kernel(const float* __restrict__ deg, const float* __restrict__ Tac,
                                  float* __restrict__ cs, int Nn) {
  int i = blockIdx.x * blockDim.x + threadIdx.x;
  if (i < Nn) {
    float dg = deg[i];
    cs[i] = rsqrtf(dg) * Tac[i] + 1.0f / dg;
  }
}

// ---------------------------------------------------------------- GEMM1
// h1[n,f] = c[n] * (x[n,:] . W1[f,:]) + b1[f]; also accumulate per-feature
// sum / sum-of-squares for batchnorm. Block = 128 nodes, 8 waves x 16 rows.
__global__ __launch_bounds__(256)
void gcn_gemm1_kernel(const float* __restrict__ x, const float* __restrict__ W1,
                      const float* __restrict__ b1, const float* __restrict__ cs,
                      float* __restrict__ h1, float* __restrict__ gsum,
                      float* __restrict__ gssq, int Nn) {
  extern __shared__ float smem[];
  float* xs   = smem;                          // 128*132
  float* w1s  = xs  + FEAT * LDS_STRIDE;       // 128*132
  float* ssum = w1s + FEAT * LDS_STRIDE;       // 128
  float* sssq = ssum + FEAT;                   // 128

  const int tid  = threadIdx.x;
  const int mblk = blockIdx.x * 128;

  if (tid < FEAT) { ssum[tid] = 0.0f; sssq[tid] = 0.0f; }

  // Stage x tile (guarded) and W1 into LDS, float4, coalesced.
  for (int idx = tid; idx < FEAT * 32; idx += 256) {
    int r  = idx >> 5;
    int c4 = (idx & 31) << 2;
    int m  = mblk + r;
    float4 v = make_float4(0.f, 0.f, 0.f, 0.f);
    if (m < Nn) v = *(const float4*)&x[(size_t)m * FEAT + c4];
    *(float4*)&xs[r * LDS_STRIDE + c4] = v;
    float4 w = *(const float4*)&W1[(size_t)r * FEAT + c4];
    *(float4*)&w1s[r * LDS_STRIDE + c4] = w;
  }
  __syncthreads();

  const int wave = tid >> 5;
  const int lane = tid & 31;
  const int lo   = lane & 15;
  const int hi   = lane >> 4;
  const int arow = wave * 16 + lo;      // A-fragment row (local)

  v8f acc[8] = {};   // 8 N-tiles of 16 features each -> 64 acc VGPRs

  for (int k0 = 0; k0 < FEAT; k0 += 4) {
    v2f a = *(const v2f*)&xs[arow * LDS_STRIDE + k0 + 2 * hi];
#pragma unroll
    for (int t = 0; t < 8; ++t) {
      v2f b = *(const v2f*)&w1s[(t * 16 + lo) * LDS_STRIDE + k0 + 2 * hi];
      acc[t] = __builtin_amdgcn_wmma_f32_16x16x4_f32(
          false, a, false, b, (short)0, acc[t], false, false);
    }
  }

  // Epilogue: scale by c[n], add b1, store h1, accumulate batchnorm stats.
  const int m0 = mblk + wave * 16;
  float csv[8];
#pragma unroll
  for (int r = 0; r < 8; ++r) {
    int m = m0 + r + 8 * hi;
    csv[r] = (m < Nn) ? cs[m] : 0.0f;
  }
#pragma unroll
  for (int t = 0; t < 8; ++t) {
    int f = t * 16 + lo;
    float b1v = b1[f];
    float ls = 0.0f, lss = 0.0f;
#pragma unroll
    for (int r = 0; r < 8; ++r) {
      int m = m0 + r + 8 * hi;
      if (m < Nn) {
        float hv = csv[r] * acc[t][r] + b1v;
        h1[(size_t)m * FEAT + f] = hv;
        ls += hv;
        lss += hv * hv;
      }
    }
    atomicAdd(&ssum[f], ls);
    atomicAdd(&sssq[f], lss);
  }
  __syncthreads();
  if (tid < FEAT) {
    atomicAdd(&gsum[tid], ssum[tid]);
    atomicAdd(&gssq[tid], sssq[tid]);
  }
}

// ---------------------------------------------------------------- stats
__global__ void gcn_stats_kernel(const float* __restrict__ gsum, const float* __restrict__ gssq,
                                 float* __restrict__ mean, float* __restrict__ rstd, int Nn) {
  int i = threadIdx.x;
  if (i < FEAT) {
    float inv_n = 1.0f / (float)Nn;
    float mu  = gsum[i] * inv_n;
    float var = gssq[i] * inv_n - mu * mu;
    mean[i] = mu;
    rstd[i] = rsqrtf(var + 1e-5f);
  }
}

// ---------------------------------------------------------------- GEMM2
// out[n,o] = c[n] * sum_h relu((h1[n,h]-mean[h])*rstd[h]) * W2[o,h] + b2[o]
__global__ __launch_bounds__(256)
void gcn_gemm2_kernel(const float* __restrict__ h1, const float* __restrict__ W2,
                      const float* __restrict__ b2, const float* __restrict__ cs,
                      const float* __restrict__ mean, const float* __restrict__ rstd,
                      float* __restrict__ out, int Nn) {
  extern __shared__ float smem[];
  float* gs  = smem;                           // 128*132 (normalized/relu'd h1)
  float* w2s = gs + FEAT * LDS_STRIDE;         // 64*132
  float* mS  = w2s + OUTF * LDS_STRIDE;        // 128
  float* rS  = mS + FEAT;                      // 128

  const int tid  = threadIdx.x;
  const int mblk = blockIdx.x * 128;

  if (tid < FEAT) { mS[tid] = mean[tid]; rS[tid] = rstd[tid]; }
  __syncthreads();

  // Stage normalized+ReLU h1 tile.
  for (int idx = tid; idx < FEAT * 32; idx += 256) {
    int r  = idx >> 5;
    int c4 = (idx & 31) << 2;
    int m  = mblk + r;
    float4 v = make_float4(0.f, 0.f, 0.f, 0.f);
    if (m < Nn) v = *(const float4*)&h1[(size_t)m * FEAT + c4];
    float4 g;
    g.x = fmaxf((v.x - mS[c4 + 0]) * rS[c4 + 0], 0.0f);
    g.y = fmaxf((v.y - mS[c4 + 1]) * rS[c4 + 1], 0.0f);
    g.z = fmaxf((v.z - mS[c4 + 2]) * rS[c4 + 2], 0.0f);
    g.w = fmaxf((v.w - mS[c4 + 3]) * rS[c4 + 3], 0.0f);
    *(float4*)&gs[r * LDS_STRIDE + c4] = g;
  }
  // Stage W2 (64 x 128).
  for (int idx = tid; idx < OUTF * 32; idx += 256) {
    int r  = idx >> 5;
    int c4 = (idx & 31) << 2;
    *(float4*)&w2s[r * LDS_STRIDE + c4] = *(const float4*)&W2[(size_t)r * FEAT + c4];
  }
  __syncthreads();

  const int wave = tid >> 5;
  const int lane = tid & 31;
  const int lo   = lane & 15;
  const int hi   = lane >> 4;
  const int arow = wave * 16 + lo;

  v8f acc[4] = {};   // 4 N-tiles cover the 64 output features

  for (int k0 = 0; k0 < FEAT; k0 += 4) {
    v2f a = *(const v2f*)&gs[arow * LDS_STRIDE + k0 + 2 * hi];
#pragma unroll
    for (int t = 0; t < 4; ++t) {
      v2f b = *(const v2f*)&w2s[(t * 16 + lo) * LDS_STRIDE + k0 + 2 * hi];
      acc[t] = __builtin_amdgcn_wmma_f32_16x16x4_f32(
          false, a, false, b, (short)0, acc[t], false, false);
    }
  }

  const int m0 = mblk + wave * 16;
  float csv[8];
#pragma unroll
  for (int r = 0; r < 8; ++r) {
    int m = m0 + r + 8 * hi;
    csv[r] = (m < Nn) ? cs[m] : 0.0f;
  }
#pragma unroll
  for (int t = 0; t < 4; ++t) {
    int o = t * 16 + lo;
    float b2v = b2[o];
#pragma unroll
    for (int r = 0; r < 8; ++r) {
      int m = m0 + r + 8 * hi;
      if (m < Nn) out[(size_t)m * OUTF + o] = csv[r] * acc[t][r] + b2v;
    }
  }
}

// ---------------------------------------------------------------- launch
extern "C" void kernel_launch(void* const* d_in, const int* in_sizes, int n_in,
                              void* d_out, int out_size, void* d_ws, size_t ws_size,
                              hipStream_t stream) {
  const float* x  = (const float*)d_in[0];
  const int*   ei = (const int*)d_in[1];
  const float* W1 = (const float*)d_in[2];
  const float* b1 = (const float*)d_in[3];
  const float* W2 = (const float*)d_in[4];
  const float* b2 = (const float*)d_in[5];
  float* out = (float*)d_out;

  const int Nn   = in_sizes[0] / FEAT;   // 100000
  const int Ecnt = in_sizes[1] / 2;      // 1600000
  const int* src = ei;
  const int* dst = ei + Ecnt;

  // Workspace layout (floats): deg[N] | T[N] | c[N] | h1[N*128] | gsum[128] |
  // gssq[128] | mean[128] | rstd[128]
  float* ws   = (float*)d_ws;
  float* deg  = ws;
  float* Tac  = ws + (size_t)Nn;
  float* cs   = ws + 2 * (size_t)Nn;
  float* h1   = ws + 3 * (size_t)Nn;
  float* gsum = h1 + (size_t)Nn * FEAT;
  float* gssq = gsum + FEAT;
  float* mean = gssq + FEAT;
  float* rstd = mean + FEAT;

  const size_t lds1 = (size_t)(2 * FEAT * LDS_STRIDE + 2 * FEAT) * sizeof(float);
  const size_t lds2 = (size_t)((FEAT + OUTF) * LDS_STRIDE + 2 * FEAT) * sizeof(float);
  // CDNA5 WGP has 320KB LDS; raise the per-kernel dynamic-LDS cap.
  (void)hipFuncSetAttribute((const void*)gcn_gemm1_kernel,
                            hipFuncAttributeMaxDynamicSharedMemorySize, (int)lds1);
  (void)hipFuncSetAttribute((const void*)gcn_gemm2_kernel,
                            hipFuncAttributeMaxDynamicSharedMemorySize, (int)lds2);

  const int tb = 256;
  gcn_init_kernel<<<(Nn + tb - 1) / tb, tb, 0, stream>>>(deg, Tac, gsum, gssq, Nn);
  gcn_deg_kernel<<<(Ecnt + tb - 1) / tb, tb, 0, stream>>>(src, dst, deg, Ecnt);
  gcn_tsum_kernel<<<(Ecnt + tb - 1) / tb, tb, 0, stream>>>(src, dst, deg, Tac, Ecnt);
  gcn_cscale_kernel<<<(Nn + tb - 1) / tb, tb, 0, stream>>>(deg, Tac, cs, Nn);

  const int mblocks = (Nn + 127) / 128;
  gcn_gemm1_kernel<<<mblocks, 256, lds1, stream>>>(x, W1, b1, cs, h1, gsum, gssq, Nn);
  gcn_stats_kernel<<<1, 128, 0, stream>>>(gsum, gssq, mean, rstd, Nn);
  gcn_gemm2_kernel<<<mblocks, 256, lds2, stream>>>(h1, W2, b2, cs, mean, rstd, out, Nn);
}